// HL_HGCNN_ABCD_dense_int3_attpool_62500364091975
// MI455X (gfx1250) — compile-verified
//
#include <hip/hip_runtime.h>
#include <hip/hip_bf16.h>

// ===========================================================================
// HL-HGCNN forward for MI455X (gfx1250), wave32 + WMMA f32 16x16x4 + TDM.
//
// Roofline: ~57 GFLOP fp32 of dense GEMM, ~1-2 GB activation traffic (xs
// panel = 92MB lives in the 192MB L2). HBM floor ~50-100us at 23.3 TB/s.
// V_WMMA_F32_16X16X4_F32 keeps full fp32 precision (the Laguerre recursion
// has cancellation) while matching that floor. Data movement uses the
// CDNA5-specific paths:
//   - TENSOR_LOAD_TO_LDS (TDM) DMAs each 128x32 A tile (strided 2D tile of
//     the ld=704 activation panel) into LDS, double-buffered, overlapped
//     with WMMA of the previous chunk; synced with s_wait_tensorcnt.
//     TDM pad feature (2 dwords per 32-dword row) gives LDS row stride 34
//     -> conflict-free ds reads (34*r mod 64 distinct for r=0..15).
//   - GLOBAL_LOAD_ASYNC_TO_LDS_B64 stages the 32x16 B weight slab straight
//     into LDS on the ASYNCcnt path.
// ===========================================================================

typedef float v2f __attribute__((ext_vector_type(2)));
typedef float v8f __attribute__((ext_vector_type(8)));
typedef unsigned int v4ui __attribute__((ext_vector_type(4)));
typedef int v4i __attribute__((ext_vector_type(4)));
typedef int v8i __attribute__((ext_vector_type(8)));
typedef int w2i __attribute__((vector_size(8)));   // matches builtin param type

#define N_NODE 16384
#define N_EDGE 32768
#define NNZ_T_ 65536
#define NNZ_S_ 131072
#define LDF 704          // final feature width, used as panel stride
#define AMAX (32768*576) // per-temp-buffer size (floats)
#define ROWPAD 34        // LDS row stride (32 data + 2 pad) for A tiles

#if __has_builtin(__builtin_amdgcn_tensor_load_to_lds)
#define HAVE_TDM 1
#else
#define HAVE_TDM 0
#endif
#if __has_builtin(__builtin_amdgcn_global_load_async_to_lds_b64)
#define HAVE_ASYNC 1
#else
#define HAVE_ASYNC 0
#endif

#if HAVE_ASYNC
// global (AS1) / LDS (AS3) typed pointers for the async-to-LDS builtin
#define GPTR64(p) ((__attribute__((address_space(1))) w2i*)(unsigned long long)(p))
#define LPTR64(p) ((__attribute__((address_space(3))) w2i*)(unsigned int)(unsigned long long)(p))
#endif

__device__ __forceinline__ float lrelu_(float v) { return v > 0.f ? v : 0.1f * v; }

// ---------------------------------------------------------------------------
// Generic fused HL GEMM:  Y[:, col0:col0+Nf] = act( sum_t T_t @ W[t] + b )
// grid = (M/128, Nf/16), block = 256 = 8 waves, one 16x16 tile per wave.
// Double-buffered LDS staging: A tile via TDM, B slab via async-to-LDS.
// ---------------------------------------------------------------------------
__global__ __launch_bounds__(256) void k_gemm_hl(
    const float* __restrict__ X0, int ldx0,
    const float* __restrict__ X1, const float* __restrict__ X2,
    const float* __restrict__ X3, int ldp,
    const float* __restrict__ W, const float* __restrict__ bias,
    float* __restrict__ Y, int ldy, int col0,
    int M, int Kd, int Nf, int nT, int fuseLrelu)
{
  __shared__ float As[2][128 * ROWPAD];
  __shared__ float Bs[2][32 * 16];

  const int tid   = threadIdx.x;
  const int lane  = tid & 31;
  const int wave  = tid >> 5;
  const int tileMwg = blockIdx.x * 128;
  const int tileN   = blockIdx.y * 16;
  const int mrow  = lane & 15;          // A row within wave tile
  const int klo   = (lane >> 4) << 1;   // 0 or 2: K pair owned by this lane
  const int bcol  = lane & 15;          // B/C column within tile

  const int kcPerT = Kd >> 5;           // 32-wide K chunks per T matrix
  const int Q      = nT * kcPerT;       // total chunks

  // ---- stage A chunk q into As[buf]: 128 rows x 32 cols, LDS stride 34 ----
  auto stageA = [&](int q, int buf) {
    const int t  = q / kcPerT;
    const int kc = (q - t * kcPerT) << 5;
    const float* X   = (t == 0) ? X0 : (t == 1) ? X1 : (t == 2) ? X2 : X3;
    const int    ldx = (t == 0) ? ldx0 : ldp;
    const float* g   = X + (size_t)tileMwg * ldx + kc;
#if HAVE_TDM
    if (wave == 0) {
      unsigned long long ga = (unsigned long long)g;
      unsigned ldsOff = (unsigned)(unsigned long long)&As[buf][0];
      v4ui g0;
      g0.x = 1u;                                   // count=1, user descriptor
      g0.y = ldsOff;                               // lds_addr
      g0.z = (unsigned)ga;                         // global_addr[31:0]
      g0.w = ((unsigned)(ga >> 32) & 0x01FFFFFFu) | (2u << 30); // ga[56:32]|type=2
      v8i g1;
      g1[0] = (2 << 16)        // data_size = 4B
            | (1 << 20)        // pad_enable
            | (4 << 22)        // pad_interval: 32 dwords
            | (1 << 25);       // pad_amount: 2 dwords
      g1[1] = (int)(32u  << 16);   // tensor_dim0[15:0]=32 (atomic_barrier=0)
      g1[2] = (int)(128u << 16);   // tensor_dim0 hi=0 | tensor_dim1[15:0]=128
      g1[3] = (int)(32u  << 16);   // tensor_dim1 hi=0 | tile_dim0=32
      g1[4] = 128;                 // tile_dim1=128, tile_dim2=0
      g1[5] = (int)(unsigned)ldx;  // tensor_dim0_stride[31:0] (elements)
      g1[6] = 0;                   // stride0 hi | stride1 lo (unused for 2D)
      g1[7] = 0;
      v4i zz4 = {0, 0, 0, 0};
      v8i zz8 = {0, 0, 0, 0, 0, 0, 0, 0};
      __builtin_amdgcn_tensor_load_to_lds(g0, g1, zz4, zz4, zz8, 0);
    }
#else
    // cooperative fallback: 256 threads, 16 floats each (8x b64)
    const int r = tid >> 1;
    const int c = (tid & 1) << 4;
    const float* src = g + (size_t)r * ldx + c;
    float* dst = &As[buf][r * ROWPAD + c];
#pragma unroll
    for (int u = 0; u < 16; u += 2)
      *(v2f*)&dst[u] = *(const v2f*)&src[u];
#endif
  };

  // ---- stage B chunk q into Bs[buf]: 32x16 slab, 2 floats per thread -----
  auto stageB = [&](int q, int buf) {
    const int t  = q / kcPerT;
    const int kc = (q - t * kcPerT) << 5;
    const float* Wt = W + (size_t)t * Kd * Nf + tileN;
    const int e  = tid * 2;
    const int kk = e >> 4, nn = e & 15;
    const float* src = &Wt[(size_t)(kc + kk) * Nf + nn];
#if HAVE_ASYNC
    __builtin_amdgcn_global_load_async_to_lds_b64(GPTR64(src), LPTR64(&Bs[buf][e]), 0, 0);
#else
    *(v2f*)&Bs[buf][e] = *(const v2f*)src;
#endif
  };

  auto waitStages = [&]() {
#if HAVE_TDM
    if (wave == 0) __builtin_amdgcn_s_wait_tensorcnt(0);
#endif
#if HAVE_ASYNC
#if __has_builtin(__builtin_amdgcn_s_wait_asynccnt)
    __builtin_amdgcn_s_wait_asynccnt(0);
#else
    asm volatile("s_wait_asynccnt 0" ::: "memory");
#endif
#endif
    __syncthreads();
  };

  v8f acc = {};

  stageA(0, 0);
  stageB(0, 0);
  waitStages();

  for (int q = 0; q < Q; ++q) {
    const int buf = q & 1;
    if (q + 1 < Q) {            // overlap next chunk's DMA with this WMMA
      stageA(q + 1, buf ^ 1);
      stageB(q + 1, buf ^ 1);
    }
    const float* Ar = &As[buf][(wave * 16 + mrow) * ROWPAD];
#pragma unroll
    for (int kk = 0; kk < 32; kk += 4) {
      v2f a = *(const v2f*)&Ar[kk + klo];
      v2f b;
      b.x = Bs[buf][(kk + klo + 0) * 16 + bcol];
      b.y = Bs[buf][(kk + klo + 1) * 16 + bcol];
      acc = __builtin_amdgcn_wmma_f32_16x16x4_f32(
          /*neg_a=*/false, a, /*neg_b=*/false, b,
          /*c_mod=*/(short)0, acc, /*reuse_a=*/false, /*reuse_b=*/false);
    }
    if (q + 1 < Q) waitStages();
  }

  const float bv    = bias[tileN + bcol];
  const int   rbase = tileMwg + wave * 16 + ((lane >> 4) << 3);
#pragma unroll
  for (int v = 0; v < 8; ++v) {
    float r = acc[v] + bv;
    if (fuseLrelu) r = lrelu_(r);
    Y[(size_t)(rbase + v) * ldy + col0 + tileN + bcol] = r;
  }
}

// ---------------------------------------------------------------------------
// Elementwise / sparse helper kernels
// ---------------------------------------------------------------------------
__global__ void k_fill(float* __restrict__ p, float v, size_t n) {
  size_t i = (size_t)blockIdx.x * blockDim.x + threadIdx.x;
  if (i < n) p[i] = v;
}

// Y (packed, ld=d) = a*X1 + b*X0, both possibly strided views
__global__ void k_axpby2(float* __restrict__ Y,
                         const float* __restrict__ X1, int ld1, float a,
                         const float* __restrict__ X0, int ld0, float b,
                         int rows, int d) {
  size_t i = (size_t)blockIdx.x * blockDim.x + threadIdx.x;
  size_t total = (size_t)rows * d;
  if (i >= total) return;
  int r = (int)(i / d), c = (int)(i % d);
  Y[i] = a * X1[(size_t)r * ld1 + c] + b * X0[(size_t)r * ld0 + c];
}

__global__ void k_scale(float* __restrict__ p, float s, size_t n) {
  size_t i = (size_t)blockIdx.x * blockDim.x + threadIdx.x;
  if (i < n) p[i] *= s;
}

// acc[ei0[e]*d + c] -= ew[e] * X[ei1[e]*ldx + c]   (Laplacian scatter term)
__global__ void k_spmv_sub(float* __restrict__ acc, int d,
                           const float* __restrict__ X, int ldx,
                           const int* __restrict__ ei0,
                           const int* __restrict__ ei1,
                           const float* __restrict__ ew, int nnz) {
  size_t i = (size_t)blockIdx.x * blockDim.x + threadIdx.x;
  size_t total = (size_t)nnz * d;
  if (i >= total) return;
  int e = (int)(i / d), c = (int)(i % d);
  float v = -ew[e] * X[(size_t)ei1[e] * ldx + c];
  unsafeAtomicAdd(&acc[(size_t)ei0[e] * d + c], v);
}

__global__ void k_deg(float* __restrict__ D, const int* __restrict__ idx, int n) {
  int i = blockIdx.x * blockDim.x + threadIdx.x;
  if (i < n) unsafeAtomicAdd(&D[idx[i]], 1.0f);
}

// mt[i[e]] -= xs[e];  mt[j[e]] += xs[e]   (incidence scatter)
__global__ void k_inc_scatter(float* __restrict__ mt, int d,
                              const float* __restrict__ XS, int ldxs,
                              const int* __restrict__ i0,
                              const int* __restrict__ i1, int E) {
  size_t i = (size_t)blockIdx.x * blockDim.x + threadIdx.x;
  size_t total = (size_t)E * d;
  if (i >= total) return;
  int e = (int)(i / d), c = (int)(i % d);
  float v = XS[(size_t)e * ldxs + c];
  unsafeAtomicAdd(&mt[(size_t)i0[e] * d + c], -v);
  unsafeAtomicAdd(&mt[(size_t)i1[e] * d + c], v);
}

__global__ void k_divrows(float* __restrict__ mt, const float* __restrict__ D,
                          int rows, int d) {
  size_t i = (size_t)blockIdx.x * blockDim.x + threadIdx.x;
  size_t total = (size_t)rows * d;
  if (i >= total) return;
  int r = (int)(i / d);
  mt[i] /= D[r];
}

// P[e*d+c] = XT[j[e]*ld + c] - XT[i[e]*ld + c]
__global__ void k_gatherdiff(float* __restrict__ P, int d,
                             const float* __restrict__ XT, int ldxt,
                             const int* __restrict__ i0,
                             const int* __restrict__ i1, int E) {
  size_t i = (size_t)blockIdx.x * blockDim.x + threadIdx.x;
  size_t total = (size_t)E * d;
  if (i >= total) return;
  int e = (int)(i / d), c = (int)(i % d);
  P[i] = XT[(size_t)i1[e] * ldxt + c] - XT[(size_t)i0[e] * ldxt + c];
}

// edge init conv: din=1 -> outer product, Nf = 32
__global__ void k_outer1(float* __restrict__ Y, int ldy,
                         const float* __restrict__ X0,
                         const float* __restrict__ A1,
                         const float* __restrict__ A2,
                         const float* __restrict__ A0,
                         const float* __restrict__ W,   // (4,1,32)
                         const float* __restrict__ b, int E) {
  size_t i = (size_t)blockIdx.x * blockDim.x + threadIdx.x;
  if (i >= (size_t)E * 32) return;
  int e = (int)(i >> 5), c = (int)(i & 31);
  float s = b[c] + X0[e] * W[c] + A1[e] * W[32 + c] +
            A2[e] * W[64 + c] + A0[e] * W[96 + c];
  Y[(size_t)e * ldy + c] = lrelu_(s);
}

// readout GEMV: r[row] = dot(X[row,:Kd], w) + b[0]; one wave per row
__global__ __launch_bounds__(256) void k_gemv(
    float* __restrict__ r, const float* __restrict__ X, int ldx, int Kd,
    const float* __restrict__ w, const float* __restrict__ b, int M) {
  int wave = threadIdx.x >> 5, lane = threadIdx.x & 31;
  int row = blockIdx.x * 8 + wave;
  if (row >= M) return;
  const float* xr = X + (size_t)row * ldx;
  float s = 0.f;
  for (int c = lane; c < Kd; c += 32) s += xr[c] * w[c];
#pragma unroll
  for (int o = 16; o > 0; o >>= 1) s += __shfl_down(s, o, 32);
  if (lane == 0) r[row] = s + b[0];
}

// final: out[b] = sum_k concat(rs_row, rt_row)[k] * W[k] + bias
__global__ __launch_bounds__(256) void k_final(
    float* __restrict__ out, const float* __restrict__ RS,
    const float* __restrict__ RT, const float* __restrict__ W,
    const float* __restrict__ bias) {
  __shared__ float red[256];
  int b = blockIdx.x, tid = threadIdx.x;
  float s = 0.f;
  for (int k = tid; k < 12288; k += 256) {
    float xv = (k < 8192) ? RS[b * 8192 + k] : RT[b * 4096 + (k - 8192)];
    s += xv * W[k];
  }
  red[tid] = s;
  __syncthreads();
  for (int o = 128; o > 0; o >>= 1) {
    if (tid < o) red[tid] += red[tid + o];
    __syncthreads();
  }
  if (tid == 0) out[b] = red[0] + bias[0];
}

// ===========================================================================
// Host orchestration
// ===========================================================================
static inline unsigned b1d(size_t total) { return (unsigned)((total + 255) / 256); }

extern "C" void kernel_launch(void* const* d_in, const int* in_sizes, int n_in,
                              void* d_out, int out_size, void* d_ws, size_t ws_size,
                              hipStream_t stream) {
  // ---- inputs (setup_inputs dict order, params flattened in insertion order)
  const float* x_t  = (const float*)d_in[0];
  const float* x_s  = (const float*)d_in[1];
  const int*   eit  = (const int*)d_in[2];   // (2, NNZ_T)
  const float* ewt  = (const float*)d_in[3];
  const int*   eis  = (const int*)d_in[4];   // (2, NNZ_S)
  const float* ews  = (const float*)d_in[5];
  const int*   inc  = (const int*)d_in[6];   // (2, E)
  const float* embW = (const float*)d_in[7];
  const float* embB = (const float*)d_in[8];
  const float* iWt  = (const float*)d_in[9];
  const float* iBt  = (const float*)d_in[10];
  const float* iWs  = (const float*)d_in[11];
  const float* iBs  = (const float*)d_in[12];
  // block conv params: index 13 + (blk*2+layer)*4 + {0:Wt,1:bt,2:Ws,3:bs}
  // msi params:        index 37 + blk*4        + {0:Wmt,1:bmt,2:Wms,3:bms}
  const float* roWt = (const float*)d_in[49];
  const float* roBt = (const float*)d_in[50];
  const float* roWs = (const float*)d_in[51];
  const float* roBs = (const float*)d_in[52];
  const float* outW = (const float*)d_in[53];
  const float* outB = (const float*)d_in[54];

  // ---- workspace layout
  float* ws  = (float*)d_ws;
  float* XE  = ws;                          // 16384*64
  float* XT  = XE + (size_t)N_NODE * 64;    // 16384*704
  float* XS  = XT + (size_t)N_NODE * LDF;   // 32768*704
  float* A1  = XS + (size_t)N_EDGE * LDF;   // 32768*576
  float* A2  = A1 + (size_t)AMAX;
  float* A0  = A2 + (size_t)AMAX;
  float* DEG = A0 + (size_t)AMAX;           // 16384
  float* RT  = DEG + N_NODE;                // 16384
  float* RS  = RT + N_NODE;                 // 32768

  auto gemm = [&](const float* X0, int ldx0, const float* X1, const float* X2,
                  const float* X3, int ldp, const float* W, const float* bias,
                  float* Y, int ldy, int col0, int M, int Kd, int Nf, int nT,
                  int lrelu) {
    dim3 g(M / 128, Nf / 16);
    k_gemm_hl<<<g, 256, 0, stream>>>(X0, ldx0, X1, X2, X3, ldp, W, bias, Y,
                                     ldy, col0, M, Kd, Nf, nT, lrelu);
  };

  // Laguerre polynomial basis: A1=T1, A2=T2, A0=T3 from x (T0)
  auto poly = [&](int n, const float* x, int ldx, int d, const int* ei,
                  const float* ew, int nnz) {
    size_t nd = (size_t)n * d, ed = (size_t)nnz * d;
    // T1 = x - L x
    k_axpby2<<<b1d(nd), 256, 0, stream>>>(A1, x, ldx, 1.f, x, ldx, 0.f, n, d);
    k_spmv_sub<<<b1d(ed), 256, 0, stream>>>(A1, d, x, ldx, ei, ei + nnz, ew, nnz);
    // T2 = (3*T1 - L T1 - T0) / 2
    k_axpby2<<<b1d(nd), 256, 0, stream>>>(A2, A1, d, 3.f, x, ldx, -1.f, n, d);
    k_spmv_sub<<<b1d(ed), 256, 0, stream>>>(A2, d, A1, d, ei, ei + nnz, ew, nnz);
    k_scale<<<b1d(nd), 256, 0, stream>>>(A2, 0.5f, nd);
    // T3 = (5*T2 - L T2 - 2*T1) / 3
    k_axpby2<<<b1d(nd), 256, 0, stream>>>(A0, A2, d, 5.f, A1, d, -2.f, n, d);
    k_spmv_sub<<<b1d(ed), 256, 0, stream>>>(A0, d, A2, d, ei, ei + nnz, ew, nnz);
    k_scale<<<b1d(nd), 256, 0, stream>>>(A0, 1.f / 3.f, nd);
  };

  // 1. node embedding: XE = lrelu(x_t @ embW + embB)
  gemm(x_t, 64, x_t, x_t, x_t, 64, embW, embB, XE, 64, 0, N_NODE, 64, 64, 1, 1);

  // 2. init node conv (Kd=64 -> 32 feats) into XT[:, 0:32]
  poly(N_NODE, XE, 64, 64, eit, ewt, NNZ_T_);
  gemm(XE, 64, A1, A2, A0, 64, iWt, iBt, XT, LDF, 0, N_NODE, 64, 32, 4, 1);

  // 3. init edge conv (din=1) into XS[:, 0:32]
  poly(N_EDGE, x_s, 1, 1, eis, ews, NNZ_S_);
  k_outer1<<<b1d((size_t)N_EDGE * 32), 256, 0, stream>>>(XS, LDF, x_s, A1, A2,
                                                         A0, iWs, iBs, N_EDGE);

  // 4. incidence degree (+1e-6)
  k_fill<<<b1d(N_NODE), 256, 0, stream>>>(DEG, 1e-6f, N_NODE);
  k_deg<<<b1d(2 * N_EDGE), 256, 0, stream>>>(DEG, inc, 2 * N_EDGE);

  // 5. dense blocks
  const int FILv[3] = {32, 64, 128};
  int d = 32;
  for (int blk = 0; blk < 3; ++blk) {
    int f = FILv[blk];
    for (int layer = 0; layer < 2; ++layer) {
      int base = 13 + (blk * 2 + layer) * 4;
      const float* Wt  = (const float*)d_in[base + 0];
      const float* bt  = (const float*)d_in[base + 1];
      const float* Wsp = (const float*)d_in[base + 2];
      const float* bsp = (const float*)d_in[base + 3];
      // node conv: reads XT[:, :d], writes XT[:, d:d+f]
      poly(N_NODE, XT, LDF, d, eit, ewt, NNZ_T_);
      gemm(XT, LDF, A1, A2, A0, d, Wt, bt, XT, LDF, d, N_NODE, d, f, 4, 1);
      // edge conv: reads XS[:, :d], writes XS[:, d:d+f]
      poly(N_EDGE, XS, LDF, d, eis, ews, NNZ_S_);
      gemm(XS, LDF, A1, A2, A0, d, Wsp, bsp, XS, LDF, d, N_EDGE, d, f, 4, 1);
      d += f;
    }
    // MSI (node<->edge interaction through incidence)
    int mb = 37 + blk * 4;
    const float* Wmt = (const float*)d_in[mb + 0];
    const float* bmt = (const float*)d_in[mb + 1];
    const float* Wms = (const float*)d_in[mb + 2];
    const float* bms = (const float*)d_in[mb + 3];
    // mt = scatter(+-xs) / D  (in A1, packed ld=d)
    k_fill<<<b1d((size_t)N_NODE * d), 256, 0, stream>>>(A1, 0.f, (size_t)N_NODE * d);
    k_inc_scatter<<<b1d((size_t)N_EDGE * d), 256, 0, stream>>>(
        A1, d, XS, LDF, inc, inc + N_EDGE, N_EDGE);
    k_divrows<<<b1d((size_t)N_NODE * d), 256, 0, stream>>>(A1, DEG, N_NODE, d);
    gemm(A1, d, A1, A1, A1, d, Wmt, bmt, XT, LDF, d, N_NODE, d, f, 1, 1);
    // diff = xt[j] - xt[i] (in A2, packed ld=d)  (reads only XT[:, :d])
    k_gatherdiff<<<b1d((size_t)N_EDGE * d), 256, 0, stream>>>(
        A2, d, XT, LDF, inc, inc + N_EDGE, N_EDGE);
    gemm(A2, d, A2, A2, A2, d, Wms, bms, XS, LDF, d, N_EDGE, d, f, 1, 1);
    d += f;
  }
  // d == 704 here

  // 6. readout GEMVs + final linear -> 4 outputs
  k_gemv<<<N_NODE / 8, 256, 0, stream>>>(RT, XT, LDF, 704, roWt, roBt, N_NODE);
  k_gemv<<<N_EDGE / 8, 256, 0, stream>>>(RS, XS, LDF, 704, roWs, roBs, N_EDGE);
  k_final<<<4, 256, 0, stream>>>((float*)d_out, RS, RT, outW, outB);
}